// MultiClassNonMaxSuppression_67791763800287
// MI455X (gfx1250) — compile-verified
//
#include <hip/hip_runtime.h>
#include <cstdint>

// ---------------------------------------------------------------------------
// Multi-class NMS for MI455X (gfx1250).
//
// Shapes (reference): B=4 images, N=4096 anchors, C=80 classes,
// K=100 detections/class, 100 final detections/image.
//
// Roofline: total input ~5.5MB -> fully L2 resident (192MB); HBM (23.3TB/s)
// is irrelevant. The cost is the serial greedy chain:
// B*C=320 problems x 100 rounds x (block argmax + 4096 IoU tests).
// Strategy:
//   * 1 workgroup (8 wave32s) per (b,c), grid = (C, B) so no int division.
//   * Boxes for image b (64KB) staged ONCE into LDS with
//     global_load_async_to_lds_b128 (ASYNCcnt path, s_wait_asynccnt), then
//     reused 100x from LDS via ds_load_b128. 64KB << 320KB WGP LDS.
//   * Scores live in registers (16/thread); argmax = register reduce +
//     wave32 shfl_xor + 8-entry cross-wave LDS combine.
//   * xyxy<->yxyx swap is a pure axis rename; IoU is invariant under it and
//     the reference swaps back at output, so we skip both swaps entirely.
// No WMMA: NMS has zero matmul content (IoU = min/max/clip/div).
// ---------------------------------------------------------------------------

#define NEG_SENT  (-1.0e9f)
#define K_PER     100
#define NMS_ITEMS 16   // 4096 anchors / 256 threads
#define TOP_ITEMS 32   // 8000 candidates / 256 threads (rounded up)

__device__ __forceinline__ void wave_argmax(float& v, int& i) {
  #pragma unroll
  for (int m = 16; m >= 1; m >>= 1) {
    float v2 = __shfl_xor(v, m, 32);
    int   i2 = __shfl_xor(i, m, 32);
    if (v2 > v || (v2 == v && i2 < i)) { v = v2; i = i2; }
  }
}

// Cross-wave combine: lane0 of each wave deposits, everyone re-reduces the
// 8 entries (uniform, broadcast-friendly LDS reads).
__device__ __forceinline__ void block_argmax(float& v, int& i,
                                             float* red_v, int* red_i) {
  wave_argmax(v, i);
  const int wave = threadIdx.x >> 5;
  const int lane = threadIdx.x & 31;
  if (lane == 0) { red_v[wave] = v; red_i[wave] = i; }
  __syncthreads();
  float bv = red_v[0]; int bi = red_i[0];
  #pragma unroll
  for (int w = 1; w < 8; ++w) {
    float v2 = red_v[w]; int i2 = red_i[w];
    if (v2 > bv || (v2 == bv && i2 < bi)) { bv = v2; bi = i2; }
  }
  __syncthreads();   // red_* reusable next iteration
  v = bv; i = bi;
}

// -------------------- Kernel 1: greedy NMS per (image, class) -------------
// grid = (C, B): blockIdx.x = class, blockIdx.y = image (no div/mod).
__global__ __launch_bounds__(256)
void nms_class_kernel(const float4* __restrict__ boxp,   // [B*N] xyxy
                      const float*  __restrict__ clsp,   // [B*N*C] logits
                      float*  __restrict__ ws_scores,    // [B*C*K]
                      float4* __restrict__ ws_boxes,     // [B*C*K]
                      int N, int C) {
  __shared__ float4 sbox[4096];        // 64KB staged boxes
  __shared__ float  red_v[8];
  __shared__ int    red_i[8];

  const int c   = blockIdx.x;
  const int b   = blockIdx.y;
  const int bc  = b * C + c;           // single s_mul + s_add
  const int tid = threadIdx.x;

  // ---- async-stage the image's boxes into LDS (CDNA5 ASYNCcnt path) ----
  const float4* gbase = boxp + (size_t)b * N;
  #pragma unroll
  for (int j = 0; j < NMS_ITEMS; ++j) {
    const int e = j * 256 + tid;
    if (e < N) {
      uint32_t lds_addr = (uint32_t)(uintptr_t)(&sbox[e]);  // low32 = LDS off
      const float4* g = gbase + e;
      asm volatile("global_load_async_to_lds_b128 %0, %1, off"
                   :: "v"(lds_addr), "v"(g) : "memory");
    }
  }

  // ---- overlap the DMA with logit gather: sigmoid + confidence threshold
  // sigmoid(x) > 0.5  <=>  x > 0
  float sc[NMS_ITEMS];
  #pragma unroll
  for (int j = 0; j < NMS_ITEMS; ++j) {
    const int n = j * 256 + tid;
    float x = (n < N) ? clsp[((size_t)b * N + n) * C + c] : -1.0f;
    sc[j] = (x > 0.0f) ? (1.0f / (1.0f + __expf(-x))) : NEG_SENT;
  }

  asm volatile("s_wait_asynccnt 0" ::: "memory");  // my wave's copies landed
  __syncthreads();                                  // everyone's copies landed

  // precompute candidate areas (registers)
  float area[NMS_ITEMS];
  #pragma unroll
  for (int j = 0; j < NMS_ITEMS; ++j) {
    const int n = j * 256 + tid;
    float4 o = (n < N) ? sbox[n] : make_float4(0.f, 0.f, 0.f, 0.f);
    area[j] = (o.z - o.x) * (o.w - o.y);
  }

  // ---- 100 greedy rounds ----
  for (int k = 0; k < K_PER; ++k) {
    float v = sc[0]; int idx = tid;
    #pragma unroll
    for (int j = 1; j < NMS_ITEMS; ++j) {
      const int n = j * 256 + tid;
      if (sc[j] > v) { v = sc[j]; idx = n; }   // ties -> lower n (j ascending)
    }
    block_argmax(v, idx, red_v, red_i);

    const float4 selb  = sbox[idx];            // LDS broadcast read
    const bool   valid = (v > NEG_SENT * 0.5f);
    if (tid == 0) {
      const size_t o = (size_t)bc * K_PER + k;
      ws_scores[o] = valid ? v : NEG_SENT;     // == jnp.where(valid, s, NEG)
      ws_boxes[o]  = selb;
    }

    // suppress (self-IoU==1 also kills the winner, matching .at[i].set(NEG))
    const float selArea = (selb.z - selb.x) * (selb.w - selb.y);
    #pragma unroll
    for (int j = 0; j < NMS_ITEMS; ++j) {
      const int n = j * 256 + tid;
      if (n < N) {
        float4 o  = sbox[n];                   // ds_load_b128, reused 100x
        float x1  = fmaxf(selb.x, o.x);
        float y1  = fmaxf(selb.y, o.y);
        float x2  = fminf(selb.z, o.z);
        float y2  = fminf(selb.w, o.w);
        float inter = fmaxf(x2 - x1, 0.0f) * fmaxf(y2 - y1, 0.0f);
        float iou   = inter / (selArea + area[j] - inter + 1e-9f);
        if (iou > 0.5f) sc[j] = NEG_SENT;
      }
    }
  }
}

// -------------------- Kernel 2: per-image top-100 merge -------------------
__global__ __launch_bounds__(256)
void topk_merge_kernel(const float*  __restrict__ ws_scores,  // [B*C*K]
                       const float4* __restrict__ ws_boxes,   // [B*C*K]
                       float* __restrict__ out_boxes,         // [B*100*4]
                       float* __restrict__ out_scores,        // [B*100]
                       float* __restrict__ out_cls,           // [B*100]
                       int*   __restrict__ out_num,           // [B]
                       int C) {
  __shared__ float red_v[8];
  __shared__ int   red_i[8];

  const int b   = blockIdx.x;
  const int tid = threadIdx.x;
  const int M   = C * K_PER;                 // 8000 candidates

  float s[TOP_ITEMS];
  #pragma unroll
  for (int j = 0; j < TOP_ITEMS; ++j) {
    const int m = j * 256 + tid;
    s[j] = (m < M) ? ws_scores[(size_t)b * M + m] : NEG_SENT;
  }

  int cnt = 0;
  for (int d = 0; d < K_PER; ++d) {
    float v = s[0]; int idx = tid;
    #pragma unroll
    for (int j = 1; j < TOP_ITEMS; ++j) {
      const int m = j * 256 + tid;
      if (s[j] > v) { v = s[j]; idx = m; }
    }
    block_argmax(v, idx, red_v, red_i);

    const bool valid = (v > NEG_SENT * 0.5f);
    if (valid) ++cnt;                         // uniform across block
    if (tid == 0) {
      const int o = b * K_PER + d;
      if (valid) {
        ((float4*)out_boxes)[o] = ws_boxes[(size_t)b * M + idx];
        out_scores[o] = v;
        out_cls[o]    = (float)(idx / K_PER); // candidate layout: c*K + k
      } else {
        ((float4*)out_boxes)[o] = make_float4(-1.f, -1.f, -1.f, -1.f);
        out_scores[o] = -1.0f;
        out_cls[o]    = -1.0f;
      }
    }
    // knock out the winner in its owner's registers (m = j*256 + t)
    if ((idx & 255) == tid) {
      const int j = idx >> 8;
      if (j < TOP_ITEMS) s[j] = NEG_SENT;
    }
  }
  if (tid == 0) out_num[b] = cnt;
}

// --------------------------- launcher -------------------------------------
extern "C" void kernel_launch(void* const* d_in, const int* in_sizes, int n_in,
                              void* d_out, int out_size, void* d_ws, size_t ws_size,
                              hipStream_t stream) {
  const float4* boxp = (const float4*)d_in[0];   // [B,N,4] float32
  const float*  clsp = (const float*)d_in[1];    // [B,N,C] float32

  // Recover shapes without hardcoding: out = B*(400+100+100+1) = 601*B.
  const int BN = in_sizes[0] / 4;
  const int B  = out_size / 601;
  const int N  = BN / B;
  const int C  = in_sizes[1] / BN;

  // Workspace: [B*C*K] scores | [B*C*K] float4 boxes  (640KB for 4/80/100).
  float*  ws_scores = (float*)d_ws;
  float4* ws_boxes  = (float4*)(ws_scores + (size_t)B * C * K_PER);

  // Output tuple, flattened in return order.
  float* out        = (float*)d_out;
  float* out_boxes  = out;                           // B*100*4
  float* out_scores = out + (size_t)B * 400;         // B*100
  float* out_cls    = out + (size_t)B * 500;         // B*100
  int*   out_num    = (int*)(out + (size_t)B * 600); // B (int32 bits)

  nms_class_kernel<<<dim3(C, B), dim3(256), 0, stream>>>(
      boxp, clsp, ws_scores, ws_boxes, N, C);
  topk_merge_kernel<<<dim3(B), dim3(256), 0, stream>>>(
      ws_scores, ws_boxes, out_boxes, out_scores, out_cls, out_num, C);
}